// NaNPool2d_83708912599745
// MI455X (gfx1250) — compile-verified
//
#include <hip/hip_runtime.h>
#include <math.h>

// Problem constants (fixed by the reference): x is (8, 64, 256, 256) fp32,
// 3x3 pool, stride 2, pad 1 -> out (8, 64, 128, 128) fp32.
#define B_   8
#define C_   64
#define H_   256
#define W_   256
#define OH_  128
#define OW_  128

// Tile geometry: one block -> 64(w) x 8(h) outputs of one channel.
#define OWT  64
#define OHT  8
#define IWT  (OWT * 2 + 1)   // 129 input cols needed (incl. halo)
#define IHT  (OHT * 2 + 1)   // 17 input rows needed (incl. halo)
#define LDSW 129             // LDS row pitch (floats); odd => bank-conflict-free

#define RTOL 1e-7f
#define ATOL 1e-8f

#if defined(__HIP_DEVICE_COMPILE__) && __has_builtin(__builtin_amdgcn_tensor_load_to_lds)
#define HAS_TDM 1
#else
#define HAS_TDM 0
#endif

#if defined(__HIP_DEVICE_COMPILE__) && __has_builtin(__builtin_amdgcn_global_load_async_to_lds_b32)
#define HAS_ASYNC_LDS 1
#else
#define HAS_ASYNC_LDS 0
#endif

typedef __attribute__((address_space(1))) void GlobalVoid;
typedef __attribute__((address_space(3))) void LdsVoid;
typedef __attribute__((address_space(1))) int  GInt;
typedef __attribute__((address_space(3))) int  LInt;

typedef unsigned int v4u __attribute__((ext_vector_type(4)));
typedef int          v8i __attribute__((ext_vector_type(8)));
typedef int          v4i __attribute__((ext_vector_type(4)));

__global__ __launch_bounds__(256) void nanpool2d_kernel(
    const float* __restrict__ x, float* __restrict__ out)
{
    __shared__ float tile[IHT * LDSW];

    const int bc  = blockIdx.z;            // flattened (b, c) channel
    const int ow0 = blockIdx.x * OWT;
    const int oh0 = blockIdx.y * OHT;
    const int ih0 = oh0 * 2 - 1;           // top halo row (may be -1 -> pad)
    const int iw0 = ow0 * 2 - 1;           // left halo col (may be -1 -> pad)

    const float* __restrict__ src = x   + (size_t)bc * (H_ * W_);
    float*       __restrict__ dst = out + (size_t)bc * (OH_ * OW_);

    const int tid = threadIdx.x;

#if HAS_TDM
    // ---- TDM staging: one 2D tile descriptor per block ----
    // Clamp the tile start into bounds; the missing left col / top row (the
    // only possible OOB cells: right/bottom halo hits exactly index 255) are
    // zero-filled in LDS. For left-clamped tiles (width 128) keep LDS pitch
    // 129 via TDM padding: pad_interval=6 (every 128 DWORDs) x 1 DWORD.
    const int ldx = (iw0 < 0) ? 1 : 0;         // shift LDS dest right by 1 col
    const int ldy = (ih0 < 0) ? 1 : 0;         // shift LDS dest down by 1 row
    const int cs  = iw0 + ldx;                 // clamped global start col
    const int rs  = ih0 + ldy;                 // clamped global start row
    const int tw  = IWT - ldx;                 // tile_dim0: 129 or 128
    const int th  = IHT - ldy;                 // tile_dim1: 17 or 16

    // Zero-fill pad cells (TDM never writes them; ds stores drain at barrier).
    if (ldx) for (int r = tid; r < IHT; r += 256) tile[r * LDSW] = 0.0f;
    if (ldy) for (int c = tid; c < IWT; c += 256) tile[c]        = 0.0f;

    if (tid < 32) {   // wave 0 issues the tensor DMA
        const unsigned long long ga =
            (unsigned long long)(uintptr_t)(src + rs * W_ + cs);
        const unsigned lbase =
            (unsigned)(uintptr_t)(LdsVoid*)&tile[ldy * LDSW + ldx];

        // ---- D# group 0: control / lds_addr / global_addr / type ----
        v4u g0;
        g0[0] = 1u;                                   // count=1 (valid), rest 0
        g0[1] = lbase;                                // lds_addr [63:32]
        g0[2] = (unsigned)(ga & 0xFFFFFFFFull);       // global_addr lo
        g0[3] = (unsigned)((ga >> 32) & 0x1FFFFFFull) // global_addr [120:96]
              | (2u << 30);                           // type=2 ("image")

        // ---- D# group 1: sizes / strides / pad ----
        const unsigned td0 = (unsigned)(W_ - cs);     // tensor_dim0 (elems)
        const unsigned td1 = (unsigned)(H_ - rs);     // tensor_dim1 (elems)
        const unsigned pad_en = (unsigned)ldx;        // pitch fix only if w=128
        v8i g1;
        g1[0] = (int)((2u << 16)                      // data_size=2 -> 4 bytes
              | (pad_en << 20)                        // pad_enable
              | ((pad_en ? 6u : 0u) << 22)            // pad_interval: 128 DW
              | (0u << 25));                          // pad_amount: 1 DW
        g1[1] = (int)((td0 & 0xFFFFu) << 16);         // tensor_dim0 lo16
        g1[2] = (int)(((td0 >> 16) & 0xFFFFu)         // tensor_dim0 hi16
              | ((td1 & 0xFFFFu) << 16));             // tensor_dim1 lo16
        g1[3] = (int)(((td1 >> 16) & 0xFFFFu)         // tensor_dim1 hi16
              | ((unsigned)tw << 16));                // tile_dim0
        g1[4] = (int)(unsigned)th;                    // tile_dim1 (tile_dim2=0)
        g1[5] = W_;                                   // tensor_dim0_stride lo32
        g1[6] = 0;                                    // stride hi / dim1_stride
        g1[7] = 0;

        v4i gz4 = {0, 0, 0, 0};                       // groups 2/3: 2D tensor
        v8i gz8 = {0, 0, 0, 0, 0, 0, 0, 0};           // VADDR4 slot: unused

        // amdgpu-toolchain 6-arg form: (g0, g1, g2, g3, g4, cpol)
        __builtin_amdgcn_tensor_load_to_lds(g0, g1, gz4, gz4, gz8, /*cpol=*/0);
#if __has_builtin(__builtin_amdgcn_s_wait_tensorcnt)
        __builtin_amdgcn_s_wait_tensorcnt(0);
#else
        asm volatile("s_wait_tensorcnt 0x0" ::: "memory");
#endif
    }
    __syncthreads();

#else  // ---- fallback staging paths ----
    for (int i = tid; i < IHT * IWT; i += 256) {
        const int r  = i / IWT;
        const int c  = i - r * IWT;
        const int ih = ih0 + r;
        const int iw = iw0 + c;
        const bool inb = ((unsigned)ih < (unsigned)H_) & ((unsigned)iw < (unsigned)W_);
#if HAS_ASYNC_LDS
        if (inb) {
            __builtin_amdgcn_global_load_async_to_lds_b32(
                (GInt*)(GlobalVoid*)(src + ih * W_ + iw),
                (LInt*)(LdsVoid*)&tile[r * LDSW + c],
                /*offset=*/0, /*cpol=*/0);
        } else {
            tile[r * LDSW + c] = 0.0f;
        }
#else
        tile[r * LDSW + c] = inb ? src[ih * W_ + iw] : 0.0f;
#endif
    }
#if HAS_ASYNC_LDS
#if __has_builtin(__builtin_amdgcn_s_wait_asynccnt)
    __builtin_amdgcn_s_wait_asynccnt(0);
#else
    asm volatile("s_wait_asynccnt 0x0" ::: "memory");
#endif
#endif
    __syncthreads();
#endif // HAS_TDM

    // ---- Each thread computes 2 outputs: (oy0, ox) and (oy0+4, ox) ----
    const int ox  = tid & 63;    // 0..63
    const int oy0 = tid >> 6;    // 0..3

    #pragma unroll
    for (int k = 0; k < 2; ++k) {
        const int oy = oy0 + k * 4;
        const int lr = oy * 2;
        const int lc = ox * 2;

        float w[9];
        float m = -INFINITY;
        #pragma unroll
        for (int dy = 0; dy < 3; ++dy) {
            #pragma unroll
            for (int dx = 0; dx < 3; ++dx) {
                float v = tile[(lr + dy) * LDSW + lc + dx];
                v = (v != v) ? -INFINITY : v;      // NaN -> -inf (reference)
                w[dy * 3 + dx] = v;
                m = fmaxf(m, v);
            }
        }

        // Tie count with jnp.isclose semantics: |v-m| <= atol + rtol*|m|,
        // plus v==m to cover (-inf == -inf) and exact ties.
        const float thr = ATOL + RTOL * fabsf(m);
        int cnt = 0;
        #pragma unroll
        for (int j = 0; j < 9; ++j) {
            const float v = w[j];
            const bool close = (v == m) || (fabsf(v - m) <= thr);
            cnt += close ? 1 : 0;
        }

        const float r = (cnt > 1) ? __builtin_nanf("") : m;
        dst[(oh0 + oy) * OW_ + (ow0 + ox)] = r;   // coalesced store
    }
}

extern "C" void kernel_launch(void* const* d_in, const int* in_sizes, int n_in,
                              void* d_out, int out_size, void* d_ws, size_t ws_size,
                              hipStream_t stream) {
    (void)in_sizes; (void)n_in; (void)out_size; (void)d_ws; (void)ws_size;
    const float* x  = (const float*)d_in[0];
    float*      out = (float*)d_out;

    dim3 block(256, 1, 1);
    dim3 grid(OW_ / OWT, OH_ / OHT, B_ * C_);   // (2, 16, 512)
    nanpool2d_kernel<<<grid, block, 0, stream>>>(x, out);
}